// GNN_oracle_51299089383938
// MI455X (gfx1250) — compile-verified
//
#include <hip/hip_runtime.h>
#include <hip/hip_bf16.h>
#include <math.h>

#define N_NODES 100000
#define N_EDGES 3200000
#define IN_F 13
#define H 32
#define H1 128
#define CLS_H 256
#define LN_EPS 1e-5f

typedef float v2f __attribute__((ext_vector_type(2)));
typedef float v4f __attribute__((ext_vector_type(4)));
typedef float v8f __attribute__((ext_vector_type(8)));

__device__ __forceinline__ float eluf(float x) { return x > 0.f ? x : __expf(x) - 1.f; }

__device__ __forceinline__ float wave_sum(float v) {
    #pragma unroll
    for (int m = 16; m >= 1; m >>= 1) v += __shfl_xor(v, m, 32);
    return v;
}

// ---------------- MLP_before: Linear(13,128)+ELU+LN -> Linear(128,32)+ELU+LN ----
// one wave per node
__global__ __launch_bounds__(256) void k_mlp_before(
    const float* __restrict__ x,
    const float* __restrict__ W1, const float* __restrict__ b1,
    const float* __restrict__ g1, const float* __restrict__ be1,
    const float* __restrict__ W2, const float* __restrict__ b2,
    const float* __restrict__ g2, const float* __restrict__ be2,
    float* __restrict__ hout)
{
    __shared__ float t[8][H1];
    const int wid = threadIdx.x >> 5, lane = threadIdx.x & 31;
    const int node = blockIdx.x * 8 + wid;
    const bool active = node < N_NODES;
    const int n = active ? node : (N_NODES - 1);

    float xv[IN_F];
    #pragma unroll
    for (int k = 0; k < IN_F; ++k) xv[k] = x[(size_t)n * IN_F + k];

    float v[4];
    #pragma unroll
    for (int u = 0; u < 4; ++u) {
        const int unit = lane + 32 * u;
        float acc = b1[unit];
        #pragma unroll
        for (int k = 0; k < IN_F; ++k) acc += xv[k] * W1[unit * IN_F + k];
        v[u] = eluf(acc);
    }
    // LayerNorm over 128 (4 vals/lane across the wave)
    float s = v[0] + v[1] + v[2] + v[3];
    const float mean = wave_sum(s) * (1.f / H1);
    float q = 0.f;
    #pragma unroll
    for (int u = 0; u < 4; ++u) { const float d = v[u] - mean; q += d * d; }
    const float inv = rsqrtf(wave_sum(q) * (1.f / H1) + LN_EPS);
    #pragma unroll
    for (int u = 0; u < 4; ++u) {
        const int unit = lane + 32 * u;
        t[wid][unit] = (v[u] - mean) * inv * g1[unit] + be1[unit];
    }
    __syncthreads();

    // Linear(128,32): lane -> output unit
    float acc = b2[lane];
    #pragma unroll 8
    for (int k = 0; k < H1; ++k) acc += t[wid][k] * W2[lane * H1 + k];
    acc = eluf(acc);
    const float m2 = wave_sum(acc) * (1.f / H);
    const float d2 = acc - m2;
    const float q2 = wave_sum(d2 * d2) * (1.f / H);
    const float y = d2 * rsqrtf(q2 + LN_EPS) * g2[lane] + be2[lane];
    if (active) hout[(size_t)node * H + lane] = y;
}

// ---------------- vectorized fill (n multiple of 4) ----------------
__global__ void k_fill4(float* __restrict__ p, float val, int n4)
{
    const int i = blockIdx.x * blockDim.x + threadIdx.x;
    if (i < n4) {
        v4f v = {val, val, val, val};
        *(v4f*)(p + (size_t)i * 4) = v;
    }
}

__global__ void k_fill(float* __restrict__ p, float val, int n)
{
    const int i = blockIdx.x * blockDim.x + threadIdx.x;
    if (i < n) p[i] = val;
}

// ---------------- edge scatter: agg[dst][f] = max(agg, h[src][f]*ew) ----------
// lane = feature, wave = edge -> coalesced 128B gather of h[src], native f32 atomic max
__global__ __launch_bounds__(256) void k_edge_max(
    const float* __restrict__ h,
    const int* __restrict__ src, const int* __restrict__ dst,
    const float* __restrict__ ew, float* __restrict__ agg)
{
    const unsigned tid = blockIdx.x * blockDim.x + threadIdx.x;
    const unsigned e = tid >> 5;
    const unsigned f = tid & 31u;
    if (e < N_EDGES) {
        const float msg = h[(size_t)src[e] * H + f] * ew[e];
        atomicMax(&agg[(size_t)dst[e] * H + f], msg);
    }
}

// ---------------- GIN combine: LN( ((1+eps)h + fix(agg)) @ W^T + b ) ----------
// One 16-node tile per wave; fp32 WMMA 16x16x4, K=32 chained in 8 steps.
// ISA 7.12.2 32-bit layouts: A 16x4 -> lanes 0-15 hold {K=0,K=1}, lanes 16-31 {K=2,K=3};
// B 4x16 mirrored (row striped across lanes); C/D: vgpr r = rows r (lanes 0-15) / r+8 (lanes 16-31).
// Bias is folded into the LayerNorm read (LN is over a@W^T + b), keeping the
// D-store loop branch-free so EXEC stays all-1s through the WMMA block.
__global__ __launch_bounds__(256) void k_gin_combine(
    const float* __restrict__ hin, const float* __restrict__ agg,
    const float* __restrict__ W, const float* __restrict__ bias,
    const float* __restrict__ g, const float* __restrict__ be,
    const float* __restrict__ epsp, int outf, float* __restrict__ hout)
{
    __shared__ float Bs[H][H + 1];
    __shared__ float As[8][16][H + 1];
    __shared__ float Ds[8][16][H + 1];
    const int NT = N_NODES / 16;  // 6250, exact
    const int wid = threadIdx.x >> 5, lane = threadIdx.x & 31;
    const float ep = 1.f + epsp[0];

    // stage B[k][j] = W[j][k], zero-padded columns for outf<32
    for (int idx = threadIdx.x; idx < H * H; idx += 256) {
        const int j = idx >> 5, k = idx & 31;
        Bs[k][j] = (j < outf) ? W[j * H + k] : 0.f;
    }

    int tile = blockIdx.x * 8 + wid;
    const bool dup = tile >= NT;      // tail waves recompute a valid tile, skip store
    if (dup) tile = NT - 1;
    const int node0 = tile * 16;

    // stage A = (1+eps)*h + (isfinite(agg)?agg:0), coalesced per row
    #pragma unroll
    for (int r = 0; r < 16; ++r) {
        const size_t n = (size_t)(node0 + r);
        float a = agg[n * H + lane];
        a = isfinite(a) ? a : 0.f;
        As[wid][r][lane] = ep * hin[n * H + lane] + a;
    }
    __syncthreads();

    const bool hiHalf = lane >= 16;
    const int row = hiHalf ? lane - 16 : lane;   // A row / B column within tile
    const int koff = hiHalf ? 2 : 0;
    const int jtiles = (outf > 16) ? 2 : 1;      // uniform

    for (int jt = 0; jt < jtiles; ++jt) {
        const int jbase = jt * 16;
        v8f c = {0.f, 0.f, 0.f, 0.f, 0.f, 0.f, 0.f, 0.f};
        #pragma unroll
        for (int kk = 0; kk < 8; ++kk) {
            const int k0 = kk * 4 + koff;
            v2f a, b;
            a.x = As[wid][row][k0];
            a.y = As[wid][row][k0 + 1];
            b.x = Bs[k0][jbase + row];
            b.y = Bs[k0 + 1][jbase + row];
            c = __builtin_amdgcn_wmma_f32_16x16x4_f32(
                    false, a, false, b, (short)0, c, false, false);
        }
        const int j = jbase + row;
        #pragma unroll
        for (int r8 = 0; r8 < 8; ++r8) {
            const int m = hiHalf ? r8 + 8 : r8;
            Ds[wid][m][j] = c[r8];                 // branch-free; bias added at LN read
        }
    }
    __syncthreads();

    // LayerNorm(outf) per row (bias folded in), write out
    const bool inF = lane < outf;
    const float bj = inF ? bias[lane] : 0.f;
    const float gg = inF ? g[lane] : 1.f;
    const float bb = inF ? be[lane] : 0.f;
    const float rinv = 1.f / (float)outf;
    for (int m = 0; m < 16; ++m) {
        const float v = inF ? (Ds[wid][m][lane] + bj) : 0.f;
        const float mean = wave_sum(v) * rinv;
        const float d = inF ? (v - mean) : 0.f;
        const float var = wave_sum(d * d) * rinv;
        const float y = d * rsqrtf(var + LN_EPS) * gg + bb;
        if (!dup && inF) hout[(size_t)(node0 + m) * outf + lane] = y;
    }
}

// ---------------- mean pool (sum, divided later) ----------------
__global__ __launch_bounds__(256) void k_pool(const float* __restrict__ emb,
                                              float* __restrict__ pooled)
{
    __shared__ float red[256];
    float loc[IN_F];
    #pragma unroll
    for (int f = 0; f < IN_F; ++f) loc[f] = 0.f;
    for (int n = blockIdx.x * blockDim.x + threadIdx.x; n < N_NODES;
         n += gridDim.x * blockDim.x) {
        #pragma unroll
        for (int f = 0; f < IN_F; ++f) loc[f] += emb[(size_t)n * IN_F + f];
    }
    for (int f = 0; f < IN_F; ++f) {
        red[threadIdx.x] = loc[f];
        __syncthreads();
        for (int s = 128; s > 0; s >>= 1) {
            if (threadIdx.x < s) red[threadIdx.x] += red[threadIdx.x + s];
            __syncthreads();
        }
        if (threadIdx.x == 0) atomicAdd(&pooled[f], red[0]);
        __syncthreads();
    }
}

// ---------------- classifier head: elu(p@Wc1^T+bc1)@Wc2^T+bc2 ----------------
__global__ __launch_bounds__(256) void k_classify(
    const float* __restrict__ pooled,
    const float* __restrict__ Wc1, const float* __restrict__ bc1,
    const float* __restrict__ Wc2, const float* __restrict__ bc2,
    float* __restrict__ out)
{
    __shared__ float red[256];
    __shared__ float p[IN_F];
    if (threadIdx.x < IN_F) p[threadIdx.x] = pooled[threadIdx.x] * (1.f / N_NODES);
    __syncthreads();
    const int j = threadIdx.x;
    float acc = bc1[j];
    #pragma unroll
    for (int k = 0; k < IN_F; ++k) acc += p[k] * Wc1[j * IN_F + k];
    red[j] = eluf(acc) * Wc2[j];
    __syncthreads();
    for (int s = 128; s > 0; s >>= 1) {
        if (j < s) red[j] += red[j + s];
        __syncthreads();
    }
    if (j == 0) out[0] = red[0] + bc2[0];
}

extern "C" void kernel_launch(void* const* d_in, const int* in_sizes, int n_in,
                              void* d_out, int out_size, void* d_ws, size_t ws_size,
                              hipStream_t stream)
{
    (void)in_sizes; (void)n_in; (void)out_size; (void)ws_size;
    const float* features = (const float*)d_in[0];
    const int*   src      = (const int*)d_in[1];
    const int*   dst      = (const int*)d_in[2];
    const float* ew       = (const float*)d_in[3];
    const float* W1  = (const float*)d_in[4];
    const float* b1  = (const float*)d_in[5];
    const float* g1  = (const float*)d_in[6];
    const float* be1 = (const float*)d_in[7];
    const float* W2  = (const float*)d_in[8];
    const float* b2  = (const float*)d_in[9];
    const float* g2  = (const float*)d_in[10];
    const float* be2 = (const float*)d_in[11];
    const float* Wg  = (const float*)d_in[12];
    const float* bg  = (const float*)d_in[13];
    const float* Wl  = (const float*)d_in[14];
    const float* bl  = (const float*)d_in[15];
    const float* eps = (const float*)d_in[16];
    const float* gln = (const float*)d_in[17];
    const float* bln = (const float*)d_in[18];
    const float* glast = (const float*)d_in[19];
    const float* blast = (const float*)d_in[20];
    const float* Wc1 = (const float*)d_in[21];
    const float* bc1 = (const float*)d_in[22];
    const float* Wc2 = (const float*)d_in[23];
    const float* bc2 = (const float*)d_in[24];
    float* out = (float*)d_out;

    float* h0     = (float*)d_ws;
    float* h1     = h0  + (size_t)N_NODES * H;
    float* agg    = h1  + (size_t)N_NODES * H;
    float* pooled = agg + (size_t)N_NODES * H;

    k_mlp_before<<<N_NODES / 8, 256, 0, stream>>>(features, W1, b1, g1, be1,
                                                  W2, b2, g2, be2, h0);

    const int n4       = (N_NODES * H) / 4;         // 3.2M floats -> 800k float4
    const int fillGrid = (n4 + 255) / 256;
    const int edgeGrid = N_EDGES / 8;               // 32 lanes per edge, 256/block
    const int combGrid = (N_NODES / 16 + 7) / 8;

    float* hc = h0;
    float* hn = h1;
    for (int i = 0; i < 4; ++i) {
        k_fill4<<<fillGrid, 256, 0, stream>>>(agg, -INFINITY, n4);
        k_edge_max<<<edgeGrid, 256, 0, stream>>>(hc, src, dst, ew, agg);
        k_gin_combine<<<combGrid, 256, 0, stream>>>(hc, agg, Wg + (size_t)i * H * H,
                                                    bg + (size_t)i * H, gln, bln,
                                                    eps + i, H, hn);
        float* tmp = hc; hc = hn; hn = tmp;
    }
    // last GIN layer -> 13-dim nodes_emb written straight into d_out+1
    k_fill4<<<fillGrid, 256, 0, stream>>>(agg, -INFINITY, n4);
    k_edge_max<<<edgeGrid, 256, 0, stream>>>(hc, src, dst, ew, agg);
    k_gin_combine<<<combGrid, 256, 0, stream>>>(hc, agg, Wl, bl, glast, blast,
                                                eps + 4, IN_F, out + 1);

    k_fill<<<1, 32, 0, stream>>>(pooled, 0.f, 16);
    k_pool<<<256, 256, 0, stream>>>(out + 1, pooled);
    k_classify<<<1, 256, 0, stream>>>(pooled, Wc1, bc1, Wc2, bc2, out);
}